// HierTree_73212012528277
// MI455X (gfx1250) — compile-verified
//
#include <hip/hip_runtime.h>

typedef __attribute__((ext_vector_type(2))) float v2f;
typedef __attribute__((ext_vector_type(8))) float v8f;

#define N_LOC   500000
#define N_CAT   5000
#define N_EDGES 2000000
#define DIM     128
#define NEG_SLOPE 0.2f

// ---------------------------------------------------------------------------
// Kernel 1: zero the segment-sum accumulator (d_ws). 640,000 floats = 160,000
// float4 stores.
// ---------------------------------------------------------------------------
__global__ void ht_zero_agg(float4* __restrict__ agg4, int n4) {
    int i = blockIdx.x * blockDim.x + threadIdx.x;
    if (i < n4) {
        float4 z; z.x = 0.f; z.y = 0.f; z.z = 0.f; z.w = 0.f;
        agg4[i] = z;
    }
}

// ---------------------------------------------------------------------------
// Kernel 2: gather + segment-sum. One wave32 per edge; lane L owns dims
// [4L, 4L+4). Row load = 32 lanes x b128 = one fully coalesced 512B row.
// Accumulate with non-returning f32 global atomics (dst fits in L2: 2.56 MB).
// ---------------------------------------------------------------------------
__global__ void ht_gather_scatter(const float* __restrict__ loc,
                                  const int*   __restrict__ esrc,
                                  const int*   __restrict__ edst,
                                  float*       __restrict__ agg) {
    int gid  = blockIdx.x * blockDim.x + threadIdx.x;
    int e    = gid >> 5;          // edge id (wave-uniform)
    int lane = gid & 31;
    if (e >= N_EDGES) return;

    int s = esrc[e];              // same addr across wave -> single request
    int d = edst[e];

    const float4 v = *(const float4*)(loc + (size_t)s * DIM + lane * 4);
    float* p = agg + (size_t)d * DIM + lane * 4;
    unsafeAtomicAdd(p + 0, v.x);  // global_atomic_add_f32, no return
    unsafeAtomicAdd(p + 1, v.y);
    unsafeAtomicAdd(p + 2, v.z);
    unsafeAtomicAdd(p + 3, v.w);
}

// ---------------------------------------------------------------------------
// Kernel 3: out = leaky_relu(agg @ W^T + b) via V_WMMA_F32_16X16X4_F32.
// One wave per 16x16 output tile. M tiles = 313 (5000 rows, tail clamped),
// N tiles = 8 (128 cols). 313 blocks x 8 waves = 2504 tiles exactly, so EXEC
// is all-1s at every WMMA; tail rows are clamped on load, predicated on store.
//
// f32 16x16x4 A layout (ISA 7.12.2): lanes 0-15 hold M=lane with K=0 (v0),
// K=1 (v1); lanes 16-31 hold K=2 (v0), K=3 (v1)  => per lane a contiguous
// float2 at column k + 2*(lane>>4). B (4x16 = W^T tile) mirrors this with
// row index n0 + (lane&15) into W[n,k].
// C/D: v[0..7] = rows m0 + v + 8*(lane>>4), col n0 + (lane&15).
// ---------------------------------------------------------------------------
__global__ void ht_gemm_bias_lrelu(const float* __restrict__ agg,
                                   const float* __restrict__ W,
                                   const float* __restrict__ bias,
                                   float*       __restrict__ out) {
    const int wave = threadIdx.x >> 5;
    const int lane = threadIdx.x & 31;
    const int tile = blockIdx.x * 8 + wave;   // 0..2503
    const int mt   = tile >> 3;               // 0..312
    const int nt   = tile & 7;                // 0..7
    const int m0   = mt * 16;
    const int n0   = nt * 16;
    const int half = lane >> 4;               // 0 or 1
    const int l    = lane & 15;

    int arow = m0 + l;
    if (arow > N_CAT - 1) arow = N_CAT - 1;   // branchless clamp (tail tile)
    const int brow = n0 + l;

    const float* aptr = agg + (size_t)arow * DIM + 2 * half;
    const float* bptr = W   + (size_t)brow * DIM + 2 * half;

    v8f c = {};
#pragma unroll
    for (int k = 0; k < DIM; k += 4) {
        v2f a = *(const v2f*)(aptr + k);
        v2f b = *(const v2f*)(bptr + k);
        // D = A(16x4) * B(4x16) + C ; 8 args: neg_a, A, neg_b, B, c_mod, C, reuse_a, reuse_b
        c = __builtin_amdgcn_wmma_f32_16x16x4_f32(false, a, false, b,
                                                  (short)0, c, false, false);
    }

    const int col = n0 + l;
    const float bv = bias[col];
#pragma unroll
    for (int v = 0; v < 8; ++v) {
        int row = m0 + v + 8 * half;
        float val = c[v] + bv;
        val = (val > 0.f) ? val : NEG_SLOPE * val;
        if (row < N_CAT) out[(size_t)row * DIM + col] = val;
    }
}

// ---------------------------------------------------------------------------
// Launcher. Inputs (dict order): loc_emb, W, b, edge_src, edge_dst.
// Output: 5000x128 f32. Scratch: agg accumulator (2.56 MB) in d_ws.
// ---------------------------------------------------------------------------
extern "C" void kernel_launch(void* const* d_in, const int* in_sizes, int n_in,
                              void* d_out, int out_size, void* d_ws, size_t ws_size,
                              hipStream_t stream) {
    const float* loc  = (const float*)d_in[0];
    const float* W    = (const float*)d_in[1];
    const float* bias = (const float*)d_in[2];
    const int*   src  = (const int*)d_in[3];
    const int*   dst  = (const int*)d_in[4];
    float* out = (float*)d_out;
    float* agg = (float*)d_ws;

    // 1) zero accumulator (must re-zero every call; harness doesn't)
    const int n4 = N_CAT * DIM / 4;                      // 160,000
    ht_zero_agg<<<(n4 + 255) / 256, 256, 0, stream>>>((float4*)agg, n4);

    // 2) gather + atomic segment-sum: one wave per edge
    const long long total = (long long)N_EDGES * 32;     // 64,000,000 lanes
    ht_gather_scatter<<<(int)(total / 256), 256, 0, stream>>>(loc, src, dst, agg);

    // 3) WMMA GEMM + bias + LeakyReLU: 2504 tiles, 8 waves/block
    ht_gemm_bias_lrelu<<<313, 256, 0, stream>>>(agg, W, bias, out);
}